// PerturbedTopK_8761733283897
// MI455X (gfx1250) — compile-verified
//
#include <hip/hip_runtime.h>

// Problem constants (match reference)
#define BB 128
#define NN 1000
#define DD 196
#define KK 48

#define SPLIT 10                     // chunks of rows per batch element
#define ROWS_PER_WG (NN / SPLIT)     // 100
#define WAVES 8
#define TPB (WAVES * 32)
#define NS 7                         // 7 slots * 32 lanes = 224 >= 196
#define STAGE_F 208                  // padded row (floats), 832 B = multiple of 16

static __device__ __forceinline__ unsigned bal32(bool p) {
#if __has_builtin(__builtin_amdgcn_ballot_w32)
  return __builtin_amdgcn_ballot_w32(p);
#else
  return (unsigned)__ballot(p ? 1 : 0);
#endif
}

// gfx1250 async global->LDS copy, 16 B per active lane. ASYNCcnt tracked.
static __device__ __forceinline__ void async_b128(unsigned lds_byte_off, const void* gaddr) {
  asm volatile("global_load_async_to_lds_b128 %0, %1, off"
               :: "v"(lds_byte_off), "v"(gaddr)
               : "memory");
}

__global__ void __launch_bounds__(TPB)
ptopk_zero_kernel(unsigned* __restrict__ p, int n) {
  int i = blockIdx.x * blockDim.x + threadIdx.x;
  if (i < n) p[i] = 0u;
}

// In-place: reinterpret accumulated u32 counts as float = count / N.
__global__ void __launch_bounds__(TPB)
ptopk_final_kernel(unsigned* __restrict__ p, int n, float invn) {
  int i = blockIdx.x * blockDim.x + threadIdx.x;
  if (i < n) {
    unsigned c = p[i];
    ((float*)p)[i] = (float)c * invn;
  }
}

__global__ void __launch_bounds__(TPB)
ptopk_kernel(const float* __restrict__ x, const float* __restrict__ noise,
             const float* __restrict__ sigmap, unsigned* __restrict__ outc) {
  __shared__ float xs[DD];
  __shared__ unsigned counts[KK * DD];
  __shared__ __align__(16) float stage[WAVES][2][STAGE_F];

  const int tid  = threadIdx.x;
  const int lane = tid & 31;
  const int wave = tid >> 5;
  const int wg   = blockIdx.x;            // 0 .. BB*SPLIT-1
  const int b    = wg / SPLIT;
  const int chunk = wg % SPLIT;
  const float sigma = *sigmap;
  const unsigned laneLT = (1u << lane) - 1u;

  for (int i = tid; i < DD; i += TPB) xs[i] = x[b * DD + i];
  for (int i = tid; i < KK * DD; i += TPB) counts[i] = 0u;
  __syncthreads();

  const float* rowbase =
      noise + (size_t)b * NN * DD + (size_t)(chunk * ROWS_PER_WG) * DD;

  // ---- issue helper: stage row r into buffer buf of this wave (2 async ops)
  auto issue_row = [&](int r, int buf) {
    const char* g0 = (const char*)(rowbase + (size_t)r * DD);  // 784 B, 16B-aligned
    unsigned l0 = (unsigned)(uintptr_t)&stage[wave][buf][0];
    async_b128(l0 + (unsigned)(lane * 16), g0 + lane * 16);          // floats 0..127
    if (lane < 17)
      async_b128(l0 + 512u + (unsigned)(lane * 16), g0 + 512 + lane * 16); // 128..195
  };

  int cur = 0;
  if (wave < ROWS_PER_WG) issue_row(wave, cur);

  for (int r = wave; r < ROWS_PER_WG; r += WAVES) {
    const int rn = r + WAVES;
    if (rn < ROWS_PER_WG) {
      issue_row(rn, cur ^ 1);
      asm volatile("s_wait_asynccnt 0x2" ::: "memory");  // row r's 2 ops done
    } else {
      asm volatile("s_wait_asynccnt 0x0" ::: "memory");
    }

    // ---- build order-preserving u32 keys for this row
    const float* st = &stage[wave][cur][0];
    unsigned key[NS];
#pragma unroll
    for (int j = 0; j < NS; ++j) {
      const int d = j * 32 + lane;
      unsigned u = 0u;
      if (d < DD) {
        float p = fmaf(st[d], sigma, xs[d]);       // x + sigma*noise
        unsigned t = __float_as_uint(p);
        u = (t & 0x80000000u) ? ~t : (t | 0x80000000u);
      }
      key[j] = u;                                   // 0 == -inf sentinel
    }

    // ---- radix bit-select of the K-th largest key (ballot-based, wave-uniform)
    unsigned prefix = 0u;
    int needed = KK;
    int groupSize = NS * 32;
    int bit = 31;
    for (; bit >= 0; --bit) {
      const unsigned hi = (prefix >> bit) | 1u;
      int c = 0;
#pragma unroll
      for (int j = 0; j < NS; ++j)
        c += __popc(bal32((key[j] >> bit) == hi));
      if (c >= needed) { prefix |= (1u << bit); groupSize = c; }
      else             { needed -= c; groupSize -= c; }
      if (groupSize == needed) break;               // take whole candidate group
    }
    const int sh = (bit < 0) ? 0 : bit;
    const unsigned pTop = prefix >> sh;

    // ---- membership: key-top > pTop definitely in; == pTop take first `needed`
    //      in ascending-d order (matches top_k tie-break: lower index first)
    bool sel[NS];
    int candBase = 0;
#pragma unroll
    for (int j = 0; j < NS; ++j) {
      const unsigned kt = key[j] >> sh;
      const bool cand  = (kt == pTop);
      const bool defIn = (kt > pTop);
      const unsigned bc = bal32(cand);
      const int myRank = candBase + __popc(bc & laneLT);
      sel[j] = defIn || (cand && (myRank < needed));
      candBase += __popc(bc);
    }

    // ---- rank among selected, ascending d (d = j*32+lane is monotone in j,lane)
    int selBase = 0;
#pragma unroll
    for (int j = 0; j < NS; ++j) {
      const unsigned bs = bal32(sel[j]);
      if (sel[j]) {
        const int kpos = selBase + __popc(bs & laneLT);
        const int d = j * 32 + lane;
        atomicAdd(&counts[kpos * DD + d], 1u);      // ds_add_u32
      }
      selBase += __popc(bs);
    }

    cur ^= 1;
  }

  __syncthreads();

  // ---- flush per-chunk integer counts into global u32 accumulator (exact,
  //      order-independent -> bitwise deterministic across replays)
  unsigned* outb = outc + (size_t)b * KK * DD;
  for (int i = tid; i < KK * DD; i += TPB) {
    const unsigned c = counts[i];
    if (c) atomicAdd(&outb[i], c);                  // global_atomic_add_u32
  }
}

extern "C" void kernel_launch(void* const* d_in, const int* in_sizes, int n_in,
                              void* d_out, int out_size, void* d_ws, size_t ws_size,
                              hipStream_t stream) {
  const float* x     = (const float*)d_in[0];
  const float* noise = (const float*)d_in[1];
  const float* sigma = (const float*)d_in[2];
  unsigned* outc = (unsigned*)d_out;

  const int eblocks = (out_size + TPB - 1) / TPB;
  ptopk_zero_kernel<<<eblocks, TPB, 0, stream>>>(outc, out_size);

  ptopk_kernel<<<BB * SPLIT, TPB, 0, stream>>>(x, noise, sigma, outc);

  ptopk_final_kernel<<<eblocks, TPB, 0, stream>>>(outc, out_size, 1.0f / (float)NN);
}